// DifferentiableAstar_30210799960599
// MI455X (gfx1250) — compile-verified
//
#include <hip/hip_runtime.h>
#include <math.h>

// Differentiable A* forward (training mode) for N=4096, Tmax=1024.
// Single persistent workgroup: 1024 threads = 32 wave32 waves on one WGP.
// State (g, open, hist, parents, path) lives in LDS; adj/weighted_adj
// (128 MB total) stay resident in the 192 MB L2 across the 1024 steps.
// The one-hot row gather (snm @ adj) is routed through V_WMMA_F32_16X16X4_F32
// (exact in f32), two WMMAs per 16-wide tile (adj row + weighted_adj row).

#define NN 4096
#define TPB 1024
#define PER 4           // NN / TPB
#define TMAX 1024       // 0.25 * N
#define INV_SQRT_N 0.015625f   // 1/sqrt(4096)

typedef __attribute__((ext_vector_type(2))) float v2f;
typedef __attribute__((ext_vector_type(8))) float v8f;

__global__ __launch_bounds__(TPB, 1)
void astar_persistent_kernel(const int* __restrict__ startp,
                             const int* __restrict__ goalp,
                             const float* __restrict__ hmap,
                             const float* __restrict__ adj,
                             const float* __restrict__ wadj,
                             float* __restrict__ out) {
    __shared__ float          s_g[NN];
    __shared__ unsigned char  s_open[NN];
    __shared__ unsigned char  s_hist[NN];
    __shared__ unsigned char  s_path[NN];
    __shared__ int            s_par[NN];
    __shared__ float          r_val[32];
    __shared__ int            r_idx[32];
    __shared__ int            s_ind;
    __shared__ float          s_gind;
    __shared__ int            s_done;
    __shared__ int            s_tf;

    const int tid  = threadIdx.x;
    const int lane = tid & 31;
    const int wid  = tid >> 5;
    const int start = startp[0];
    const int goal  = goalp[0];

    // ---- init state (replicates: zero diag of weighted_adj, inf->0, g0 = wadj[start]) ----
    for (int k = 0; k < PER; ++k) {
        int j = tid + k * TPB;
        float w0 = wadj[(size_t)start * NN + j];
        if (isinf(w0)) w0 = 0.0f;
        s_g[j]    = (j == start) ? 0.0f : w0;
        s_open[j] = 0;
        s_hist[j] = 0;
        s_path[j] = 0;
        s_par[j]  = goal;
    }
    __syncthreads();
    if (tid == 0) {
        s_open[start] = 1;
        s_path[goal]  = 1;   // path0 = one-hot(goal)
        s_done = 0;
        s_tf   = TMAX - 1;
    }

    // ---- main sequential A* loop ----
    for (int t = 0; t < TMAX; ++t) {
        __syncthreads();
        if (s_done) break;

        // phase 1: argmax of f_exp = exp(-f/sqrt(N)) * open  (first-index tie-break)
        float bestv = -1.0f;
        int   besti = 0x7fffffff;
        for (int k = 0; k < PER; ++k) {
            int j = tid + k * TPB;
            float f = 0.5f * (s_g[j] + hmap[j]);
            float v = s_open[j] ? expf(-f * INV_SQRT_N) : 0.0f;
            if (v > bestv || (v == bestv && j < besti)) { bestv = v; besti = j; }
        }
        for (int off = 16; off > 0; off >>= 1) {
            float ov = __shfl_down(bestv, off, 32);
            int   oi = __shfl_down(besti, off, 32);
            if (ov > bestv || (ov == bestv && oi < besti)) { bestv = ov; besti = oi; }
        }
        if (lane == 0) { r_val[wid] = bestv; r_idx[wid] = besti; }
        __syncthreads();
        if (tid < 32) {
            bestv = r_val[tid]; besti = r_idx[tid];
            for (int off = 16; off > 0; off >>= 1) {
                float ov = __shfl_down(bestv, off, 32);
                int   oi = __shfl_down(besti, off, 32);
                if (ov > bestv || (ov == bestv && oi < besti)) { bestv = ov; besti = oi; }
            }
            if (tid == 0) {
                int ind = besti;
                s_ind  = ind;
                s_gind = s_g[ind];
                s_open[ind] = 0;   // open2 = open - snm
                s_hist[ind] = 1;   // hist2 = hist + snm
                s_tf = t;          // t_final tracks t while !done
                if (ind == goal) s_done = 1;   // freeze AFTER this step's updates
            }
        }
        __syncthreads();

        // phase 2: neighbor = (snm @ adj) masked; relax g/parents/open.
        // One-hot matvec via V_WMMA_F32_16X16X4_F32 (exact in f32).
        const int ind = s_ind;
        const float gi = s_gind;
        const int kb = ind >> 2;        // 4-row K-block containing ind
        const int k0 = ind & 3;

        // A (16x4 f32): row 0 = one-hot at k0. Lanes 0-15 hold K0,K1; lanes 16-31 hold K2,K3.
        v2f a; a[0] = 0.0f; a[1] = 0.0f;
        const int src_lane = (k0 < 2) ? 0 : 16;
        if (lane == src_lane) {
            if (k0 & 1) a[1] = 1.0f; else a[0] = 1.0f;
        }

        const float* adjrow = adj  + (size_t)(kb * 4) * NN;
        const float* wrow   = wadj + (size_t)(kb * 4) * NN;
        const int half = lane >> 4;     // B layout mirrors A: VGPR0 = K{0|2}, VGPR1 = K{1|3}
        const int cL   = lane & 15;
        const int r0   = half * 2;

        for (int tt = 0; tt < 8; ++tt) {
            const int tile = wid * 8 + tt;
            const int col  = tile * 16 + cL;
            v2f bA, bW;
            bA[0] = adjrow[(size_t)r0 * NN + col];
            bA[1] = adjrow[(size_t)(r0 + 1) * NN + col];
            bW[0] = wrow[(size_t)r0 * NN + col];
            bW[1] = wrow[(size_t)(r0 + 1) * NN + col];

            v8f c  = {};
            v8f cw = {};
            // 8 args: (neg_a, A, neg_b, B, c_mod, C, reuse_a, reuse_b)
            c  = __builtin_amdgcn_wmma_f32_16x16x4_f32(false, a, false, bA, (short)0, c,  false, false);
            cw = __builtin_amdgcn_wmma_f32_16x16x4_f32(false, a, false, bW, (short)0, cw, false, false);

            if (lane < 16) {            // D row 0 (M=0) lives in lanes 0-15, VGPR0
                const int j = tile * 16 + lane;
                const float av = c[0];   // adj[ind][j]
                const float wv = cw[0];  // weighted_adj[ind][j]
                const float om = (float)s_open[j];
                const float hm = (float)s_hist[j];
                const float mask = (1.0f - om) * (1.0f - hm);
                const float nb = av * mask;                 // neighbor
                const float g2 = gi + wv;                   // tentative g through ind
                const float gj = s_g[j];
                const float idxv = (mask + om * (gj > g2 ? 1.0f : 0.0f)) * nb;
                if (idxv != 0.0f) {                         // idxv is exactly 0 or 1
                    s_g[j]    = g2 * idxv + gj * (1.0f - idxv);
                    s_par[j]  = ind;
                    s_open[j] = 1;
                }
            }
        }
    }

    // ---- backtrack: loc = parents[goal]; repeat t_final times ----
    __syncthreads();
    if (tid == 0) {
        int loc = s_par[goal];
        const int tf = s_tf;
        for (int i = 0; i < tf; ++i) {
            s_path[loc] = 1;
            loc = s_par[loc];
        }
    }
    __syncthreads();

    // ---- outputs: [hist (N), path_maps (N)] as float32 ----
    for (int k = 0; k < PER; ++k) {
        int j = tid + k * TPB;
        out[j]      = (float)s_hist[j];
        out[NN + j] = (float)s_path[j];
    }
}

extern "C" void kernel_launch(void* const* d_in, const int* in_sizes, int n_in,
                              void* d_out, int out_size, void* d_ws, size_t ws_size,
                              hipStream_t stream) {
    (void)in_sizes; (void)n_in; (void)out_size; (void)d_ws; (void)ws_size;
    const int*   startp = (const int*)d_in[0];
    const int*   goalp  = (const int*)d_in[1];
    const float* hmap   = (const float*)d_in[2];
    // d_in[3] = nodes (N,2) — unused by the forward pass
    const float* adj    = (const float*)d_in[4];
    const float* wadj   = (const float*)d_in[5];
    float* out = (float*)d_out;

    astar_persistent_kernel<<<dim3(1), dim3(TPB), 0, stream>>>(
        startp, goalp, hmap, adj, wadj, out);
}